// GMMLoss_37477884625577
// MI455X (gfx1250) — compile-verified
//
#include <hip/hip_runtime.h>
#include <math.h>

// ---------------- problem constants ----------------
#define KC 8      // mixture components
#define DC 32     // feature dim
#define W_ENERGY 0.1f
#define W_COV    0.005f
#define COV_EPS_F 1.0e-6f
#define EPS_F     1.0e-9f

// ---------------- workspace layout (floats) ----------------
#define OFF_SG    0                       // [8]   sum_gamma
#define OFF_GZ    8                       // [8][32] sum gamma*z
#define OFF_M2    264                     // [8][32][32] second moment
#define OFF_CINV  8456                    // [8][32][32] inverse covariance
#define OFF_MU    16648                   // [8][32]
#define OFF_COEF  16904                   // [8]  phi / (det 2pi cov)^(1/4)
#define OFF_CDIAG 16912                   // [1]
#define OFF_ESUM  16913                   // [1]
#define WS_FLOATS 16914

typedef __attribute__((ext_vector_type(2))) float v2f;
typedef __attribute__((ext_vector_type(8))) float v8f;
typedef __attribute__((ext_vector_type(4))) unsigned int v4u;
typedef __attribute__((ext_vector_type(8))) int v8i;
typedef __attribute__((ext_vector_type(4))) int v4i;

#if defined(__gfx1250__) && __has_builtin(__builtin_amdgcn_tensor_load_to_lds)
#define HAVE_TDM 1
#else
#define HAVE_TDM 0
#endif

__device__ __forceinline__ v8f wmma_f32_k4(v2f a, v2f b, v8f c) {
    // D(16x16,f32) = A(16x4,f32) * B(4x16,f32) + C
    return __builtin_amdgcn_wmma_f32_16x16x4_f32(
        /*neg_a=*/false, a, /*neg_b=*/false, b,
        /*c_mod=*/(short)0, c, /*reuse_a=*/false, /*reuse_b=*/false);
}

#if HAVE_TDM
// Issue one TDM 2D tile load: rows x rowLen f32 elements, contiguous rows of
// rowLen, from global 'gptr' into LDS byte offset 'lds_off'. Rows beyond
// 'tensorRows' are OOB and written as zero by the TDM (replaces tail guards).
__device__ __forceinline__ void tdm_load_tile_f32(const float* gptr, unsigned lds_off,
                                                  unsigned rowLen, unsigned rows,
                                                  unsigned tensorRows) {
    unsigned long long ga = (unsigned long long)(size_t)gptr;
    v4u g0;
    g0[0] = 1u;                                            // count=1 user descriptor
    g0[1] = lds_off;                                       // lds_addr (bytes)
    g0[2] = (unsigned)(ga & 0xFFFFFFFFu);                  // global_addr[31:0]
    g0[3] = (unsigned)((ga >> 32) & 0x01FFFFFFu) | (2u << 30);  // addr[56:32] | type=2
    v8i g1;
    g1[0] = (int)(2u << 16);                               // data_size=2 (4B), mask=0
    g1[1] = (int)((rowLen & 0xFFFFu) << 16);               // tensor_dim0[15:0]
    g1[2] = (int)(((rowLen >> 16) & 0xFFFFu) |
                  ((tensorRows & 0xFFFFu) << 16));         // tdim0 hi | tensor_dim1 lo
    g1[3] = (int)(((tensorRows >> 16) & 0xFFFFu) |
                  ((rowLen & 0xFFFFu) << 16));             // tensor_dim1 hi | tile_dim0
    g1[4] = (int)(rows & 0xFFFFu);                         // tile_dim1 (tile_dim2=0)
    g1[5] = (int)rowLen;                                   // tensor_dim0_stride lo32
    g1[6] = 0;
    g1[7] = 0;
    v4i gz = {0, 0, 0, 0};
#if defined(__clang_major__) && (__clang_major__ >= 23)
    v8i gz8 = {0, 0, 0, 0, 0, 0, 0, 0};
    __builtin_amdgcn_tensor_load_to_lds(g0, g1, gz, gz, gz8, 0);
#else
    __builtin_amdgcn_tensor_load_to_lds(g0, g1, gz, gz, 0);
#endif
}
#endif

__device__ __forceinline__ unsigned lds_offset_of(const void* p) {
    // flat->LDS addrspacecast lowers to low-32-bit truncation on AMDGPU
    return (unsigned)(size_t)p;
}

// ---------------- kernel 0: zero workspace ----------------
__global__ void gmm_init(float* __restrict__ ws) {
    int i = blockIdx.x * 256 + threadIdx.x;
    if (i < WS_FLOATS) ws[i] = 0.0f;
}

// ---------------- kernel 1: weighted statistics (WMMA pass 1) ----------------
// M2[k,d,e] = sum_n gamma[n,k] * z[n,d] * z[n,e]   via V_WMMA_F32_16X16X4_F32
// LDS tiles are double-buffered; the TDM streams tile t+1 while waves compute
// tile t (steady state: s_wait_tensorcnt <= 2 retires only the older pair).
__global__ void __launch_bounds__(256)
gmm_stats(const float* __restrict__ z, const float* __restrict__ g,
          float* __restrict__ ws, int N) {
    __shared__ float zt[2][64 * 32];   // 2 x 8KB z tiles
    __shared__ float gt[2][64 * 8];    // 2 x 2KB gamma tiles

    const int tid  = threadIdx.x;
    const int lane = tid & 31;
    const int k    = tid >> 5;      // wave id == component (8 waves)
    const int half = lane >> 4;     // 0: lanes 0-15, 1: lanes 16-31
    const int lr   = lane & 15;

    v8f acc00 = {}, acc01 = {}, acc10 = {}, acc11 = {};  // (d0,e0)(d0,e16)(d16,e0)(d16,e16)
    float sgacc = 0.0f, gzacc = 0.0f;

    const int numTiles = (N + 63) >> 6;
    const int stride   = gridDim.x;
    int buf = 0;

#if HAVE_TDM
    if (k == 0 && (int)blockIdx.x < numTiles) {    // preload first tile
        const int sBase = (int)blockIdx.x << 6;
        tdm_load_tile_f32(z + (size_t)sBase * 32, lds_offset_of(zt[0]),
                          32u, 64u, (unsigned)(N - sBase));
        tdm_load_tile_f32(g + (size_t)sBase * 8, lds_offset_of(gt[0]),
                          8u, 64u, (unsigned)(N - sBase));
    }
#endif

    for (int tile = blockIdx.x; tile < numTiles; tile += stride) {
        __syncthreads();  // all consumers of buf^1 (previous iteration) done
#if HAVE_TDM
        if (k == 0) {
            const int nextTile = tile + stride;
            if (nextTile < numTiles) {             // issue-ahead into other buffer
                const int sNext = nextTile << 6;
                tdm_load_tile_f32(z + (size_t)sNext * 32, lds_offset_of(zt[buf ^ 1]),
                                  32u, 64u, (unsigned)(N - sNext));
                tdm_load_tile_f32(g + (size_t)sNext * 8, lds_offset_of(gt[buf ^ 1]),
                                  8u, 64u, (unsigned)(N - sNext));
                __builtin_amdgcn_s_wait_tensorcnt(2);   // current tile's pair retired
            } else {
                __builtin_amdgcn_s_wait_tensorcnt(0);
            }
        }
#else
        {   // fallback: stage current tile directly
            const int sBase = tile << 6;
            #pragma unroll
            for (int t = 0; t < 2; ++t) {
                int idx = tid + (t << 8);
                int s = sBase + (idx >> 3);
                float4 v = make_float4(0.f, 0.f, 0.f, 0.f);
                if (s < N) v = *(const float4*)(z + (size_t)s * 32 + ((idx & 7) << 2));
                ((float4*)zt[buf])[idx] = v;
            }
            if (tid < 128) {
                int s = sBase + (tid >> 1);
                float4 v = make_float4(0.f, 0.f, 0.f, 0.f);
                if (s < N) v = *(const float4*)(g + (size_t)s * 8 + ((tid & 1) << 2));
                ((float4*)gt[buf])[tid] = v;
            }
        }
#endif
        __syncthreads();

        const float* zb = zt[buf];
        const float* gb = gt[buf];
        // 16 steps x 4 samples: A = gamma .* z (d rows), B = z (e cols)
        #pragma unroll 4
        for (int s0 = 0; s0 < 64; s0 += 4) {
            int sA = s0 + 2 * half;                 // K mapping: vgpr0 <-> {s0,s0+2}, vgpr1 <-> {s0+1,s0+3}
            float g0  = gb[sA * 8 + k];
            float g1  = gb[(sA + 1) * 8 + k];
            float z00 = zb[sA * 32 + lr];           // d/e in [0,16)
            float z01 = zb[(sA + 1) * 32 + lr];
            float z10 = zb[sA * 32 + 16 + lr];      // d/e in [16,32)
            float z11 = zb[(sA + 1) * 32 + 16 + lr];
            v2f b0 = { z00, z01 };
            v2f b1 = { z10, z11 };
            v2f a0 = { g0 * z00, g1 * z01 };
            v2f a1 = { g0 * z10, g1 * z11 };
            acc00 = wmma_f32_k4(a0, b0, acc00);
            acc01 = wmma_f32_k4(a0, b1, acc01);
            acc10 = wmma_f32_k4(a1, b0, acc10);
            acc11 = wmma_f32_k4(a1, b1, acc11);
        }
        // first-moment + mass accumulators (VALU, small)
        for (int s = 0; s < 64; ++s) {
            float gv = gb[s * 8 + k];
            sgacc += gv;                        // same value in every lane; lane 0 commits
            gzacc += gv * zb[s * 32 + lane];    // lane d accumulates sum gamma*z_d
        }
        buf ^= 1;
    }

    if (lane == 0) atomicAdd(ws + OFF_SG + k, sgacc);
    atomicAdd(ws + OFF_GZ + k * 32 + lane, gzacc);
    float* m2 = ws + OFF_M2 + k * 1024;
    #pragma unroll
    for (int j = 0; j < 8; ++j) {              // C layout: vgpr j -> M = j + 8*half, N = lr
        int dr = j + 8 * half;
        atomicAdd(m2 + (dr)*32      + lr,        acc00[j]);
        atomicAdd(m2 + (dr)*32      + 16 + lr,   acc01[j]);
        atomicAdd(m2 + (16 + dr)*32 + lr,        acc10[j]);
        atomicAdd(m2 + (16 + dr)*32 + 16 + lr,   acc11[j]);
    }
}

// ---------------- kernel 2: build cov, invert (in-place Gauss-Jordan), det terms ----------------
__global__ void __launch_bounds__(256)
gmm_finalize(float* __restrict__ ws, int N) {
    __shared__ float A[KC][DC][DC];    // 32 KB: cov -> inverse in place
    __shared__ float muS[KC][DC];
    __shared__ float cdiagS;

    const int tid = threadIdx.x;
    const int c   = tid & 31;          // lane = matrix column
    const int k   = tid >> 5;          // wave = component
    if (tid == 0) cdiagS = 0.0f;

    const float sg = ws[OFF_SG + k];
    const float mu = ws[OFF_GZ + k * 32 + c] / sg;
    muS[k][c] = mu;
    __syncthreads();

    // cov = M2/sg - mu mu^T + (COV_EPS + EPS) I
    const float epsd = COV_EPS_F + EPS_F;
    for (int d = 0; d < DC; ++d) {
        float cov = ws[OFF_M2 + k * 1024 + d * 32 + c] / sg - muS[k][d] * mu;
        if (d == c) cov += epsd;
        A[k][d][c] = cov;
    }
    __syncthreads();

    // cov_diag = sum_k sum_d 1/cov[k,d,d]
    float rd = 1.0f / A[k][c][c];
    #pragma unroll
    for (int m = 1; m < 32; m <<= 1) rd += __shfl_xor(rd, m, 32);
    if (c == 0) atomicAdd(&cdiagS, rd);

    // in-place Gauss-Jordan inverse (SPD, no pivoting) + log-det
    float logdet = 0.0f;
    for (int p = 0; p < DC; ++p) {
        float piv = A[k][p][p];
        logdet += logf(piv);
        __syncthreads();
        float rp = (c == p) ? 1.0f : A[k][p][c];
        rp /= piv;
        A[k][p][c] = rp;
        __syncthreads();
        for (int r = 0; r < DC; ++r) {
            if (r == p) continue;
            float f = A[k][r][p];                 // read before lane p overwrites (in-order wave)
            float v = (c == p) ? 0.0f : A[k][r][c];
            A[k][r][c] = v - f * rp;
        }
        __syncthreads();
    }

    for (int d = 0; d < DC; ++d)
        ws[OFF_CINV + k * 1024 + d * 32 + c] = A[k][d][c];
    ws[OFF_MU + k * 32 + c] = mu;
    if (c == 0) {
        float phi = sg / (float)N;
        const float LOG2PI = 1.8378770664093453f;
        // ref: divide by sqrt(prod diag chol(2pi*cov)) == (det 2pi*cov)^(1/4)
        ws[OFF_COEF + k] = phi * expf(-0.25f * (32.0f * LOG2PI + logdet));
    }
    __syncthreads();
    if (tid == 0) ws[OFF_CDIAG] = cdiagS;
}

// ---------------- kernel 3: sample energy (WMMA pass 2) ----------------
// Q = (z - mu) * Cinv per 16-sample tile, quad = <(z-mu), Q>, mix over K, -log, mean
__global__ void __launch_bounds__(256)
gmm_energy(const float* __restrict__ z, float* __restrict__ ws, int N) {
    __shared__ float CinvS[KC * DC * DC];   // 32 KB
    __shared__ float muS[KC * DC];
    __shared__ float coefS[KC];
    __shared__ float zt[2][16 * 32];        // double-buffered z tiles
    __shared__ float S[16];

    const int tid  = threadIdx.x;
    const int lane = tid & 31;
    const int k    = tid >> 5;
    const int half = lane >> 4;
    const int lr   = lane & 15;

    for (int i = tid; i < KC * DC * DC; i += 256) CinvS[i] = ws[OFF_CINV + i];
    for (int i = tid; i < KC * DC; i += 256)      muS[i]   = ws[OFF_MU + i];
    if (tid < KC) coefS[tid] = ws[OFF_COEF + tid];

    const float* Ck = CinvS + k * DC * DC;
    const float* mk = muS + k * DC;

    float esum = 0.0f;
    const int numTiles = (N + 15) >> 4;
    const int stride   = gridDim.x;
    int buf = 0;

#if HAVE_TDM
    if (k == 0 && (int)blockIdx.x < numTiles) {
        const int sBase = (int)blockIdx.x << 4;
        tdm_load_tile_f32(z + (size_t)sBase * 32, lds_offset_of(zt[0]),
                          32u, 16u, (unsigned)(N - sBase));
    }
#endif
    __syncthreads();   // CinvS/muS staged

    for (int tile = blockIdx.x; tile < numTiles; tile += stride) {
        const int sBase = tile << 4;
#if HAVE_TDM
        if (k == 0) {
            const int nextTile = tile + stride;
            if (nextTile < numTiles) {
                const int sNext = nextTile << 4;
                tdm_load_tile_f32(z + (size_t)sNext * 32, lds_offset_of(zt[buf ^ 1]),
                                  32u, 16u, (unsigned)(N - sNext));
                __builtin_amdgcn_s_wait_tensorcnt(1);   // current tile retired
            } else {
                __builtin_amdgcn_s_wait_tensorcnt(0);
            }
        }
#else
        if (tid < 128) {                         // stage 16x32 z tile
            int s = sBase + (tid >> 3);
            float4 v = make_float4(0.f, 0.f, 0.f, 0.f);
            if (s < N) v = *(const float4*)(z + (size_t)s * 32 + ((tid & 7) << 2));
            ((float4*)zt[buf])[tid] = v;
        }
#endif
        if (tid < 16) S[tid] = 0.0f;
        __syncthreads();

        const float* zb = zt[buf];
        v8f acc0 = {}, acc1 = {};                // Q cols [0,16) and [16,32)
        #pragma unroll
        for (int e0 = 0; e0 < DC; e0 += 4) {     // K-dim loop in chunks of 4
            int e = e0 + 2 * half;
            v2f a  = { zb[lr * 32 + e]     - mk[e],
                       zb[lr * 32 + e + 1] - mk[e + 1] };      // A: samples x e
            v2f b0 = { Ck[e * 32 + lr],        Ck[(e + 1) * 32 + lr] };
            v2f b1 = { Ck[e * 32 + 16 + lr],   Ck[(e + 1) * 32 + 16 + lr] };
            acc0 = wmma_f32_k4(a, b0, acc0);
            acc1 = wmma_f32_k4(a, b1, acc1);
        }

        const float coef = coefS[k];
        #pragma unroll
        for (int j = 0; j < 8; ++j) {            // C layout: vgpr j -> sample j + 8*half, d = lr (+16)
            int sj = j + 8 * half;
            float p = acc0[j] * (zb[sj * 32 + lr]      - mk[lr])
                    + acc1[j] * (zb[sj * 32 + 16 + lr] - mk[16 + lr]);
            p += __shfl_xor(p, 1, 32);
            p += __shfl_xor(p, 2, 32);
            p += __shfl_xor(p, 4, 32);
            p += __shfl_xor(p, 8, 32);           // quad form summed within 16-lane half
            if (lr == j) atomicAdd(&S[sj], coef * expf(-0.5f * p));
        }
        __syncthreads();
        if (tid < 16) {
            int n = sBase + tid;
            if (n < N) esum += -logf(S[tid] + COV_EPS_F);
        }
        __syncthreads();   // S consumed; next iteration may reset/overwrite
        buf ^= 1;
    }

    if (k == 0) {                                // wave 0 lanes 0..15 hold partial sums
        esum += __shfl_xor(esum, 1, 32);
        esum += __shfl_xor(esum, 2, 32);
        esum += __shfl_xor(esum, 4, 32);
        esum += __shfl_xor(esum, 8, 32);
        if (lane == 0) atomicAdd(ws + OFF_ESUM, esum);
    }
}

// ---------------- kernel 4: scalar output ----------------
__global__ void gmm_out(const float* __restrict__ ws, float* __restrict__ out, int N) {
    float se = ws[OFF_ESUM] / (float)N;
    if (!isfinite(se)) se = 0.0f;
    float cd = ws[OFF_CDIAG];
    if (!isfinite(cd)) cd = 0.0f;
    out[0] = W_ENERGY * se + W_COV * cd;
}

extern "C" void kernel_launch(void* const* d_in, const int* in_sizes, int n_in,
                              void* d_out, int out_size, void* d_ws, size_t ws_size,
                              hipStream_t stream) {
    const float* z = (const float*)d_in[0];
    const float* g = (const float*)d_in[1];
    float* out = (float*)d_out;
    float* ws  = (float*)d_ws;
    const int N = in_sizes[0] / DC;   // 262144 samples

    gmm_init<<<(WS_FLOATS + 255) / 256, 256, 0, stream>>>(ws);
    gmm_stats<<<512, 256, 0, stream>>>(z, g, ws, N);
    gmm_finalize<<<1, 256, 0, stream>>>(ws, N);
    gmm_energy<<<512, 256, 0, stream>>>(z, ws, N);
    gmm_out<<<1, 1, 0, stream>>>(ws, out, N);
}